// CX_Model_19636590478129
// MI455X (gfx1250) — compile-verified
//
#include <hip/hip_runtime.h>
#include <hip/hip_bf16.h>

// ---------------------------------------------------------------------------
// NNConv GNN edge scorer for MI455X (gfx1250, wave32).
//
// Key simplification: edge_attr == ones, so the per-edge weight matrix
//   W = relu(nn_w1 + nn_b1) @ nn_w2 + nn_b2   (32x32)
// is identical for every edge. segment_sum is linear, so
//   agg = scatter_add(g[src] at dst),  g = h @ W   (node-level GEMM).
//
// Dense node GEMMs use V_WMMA_F32_16X16X4_F32 (full f32 precision, 2-VGPR
// A/B fragments, 8-VGPR accumulator). Edge scatter uses global f32 atomics.
// Edge MLP (32x8, 8x1) is below WMMA tile size and gather-bound -> VALU.
// ---------------------------------------------------------------------------

typedef __attribute__((ext_vector_type(2))) float v2f;
typedef __attribute__((ext_vector_type(8))) float v8f;

#define H_DIM 32
#define IN_FEAT 64

// ---------------------------------------------------------------------------
// Kernel 1: W[32x32] = relu(nn_w1[0,:] + nn_b1) @ nn_w2 + nn_b2
// nn_w2 is [32, 1024] row-major; W stored row-major [d][o] = W[d*32+o].
// ---------------------------------------------------------------------------
__global__ __launch_bounds__(256) void wedge_kernel(
    const float* __restrict__ nn_w1, const float* __restrict__ nn_b1,
    const float* __restrict__ nn_w2, const float* __restrict__ nn_b2,
    float* __restrict__ W)
{
    __shared__ float t[H_DIM];
    int tid = threadIdx.x;
    if (tid < H_DIM) t[tid] = fmaxf(nn_w1[tid] + nn_b1[tid], 0.0f);
    __syncthreads();
    for (int p = tid; p < H_DIM * H_DIM; p += 256) {
        float acc = nn_b2[p];
        #pragma unroll
        for (int j = 0; j < H_DIM; ++j)
            acc = fmaf(t[j], nn_w2[j * (H_DIM * H_DIM) + p], acc);
        W[p] = acc;
    }
}

// ---------------------------------------------------------------------------
// Kernel 2: per-wave tile of 16 nodes.
//   h    = relu(x_tile[16x64] @ lin0_w[64x32] + lin0_b)      (WMMA, K=64)
//   g    = h @ W[32x32]                                      (WMMA, K=32)
//   out  = h @ conv_root[32x32] + conv_bias                  (WMMA, K=32)
// A-fragment layout (f32 16x4): lane L holds row m=L&15, K = 2*(L>>4)+{0,1}.
// B-fragment layout (f32 4x16): lane L holds col n=L&15, K = 2*(L>>4)+{0,1}.
// C/D layout: VGPR v -> row v + 8*(L>>4), col L&15.
// ---------------------------------------------------------------------------
__global__ __launch_bounds__(32) void node_kernel(
    const float* __restrict__ x,         // [N,64]
    const float* __restrict__ lin0_w,    // [64,32]
    const float* __restrict__ lin0_b,    // [32]
    const float* __restrict__ W,         // [32,32] (ws)
    const float* __restrict__ conv_root, // [32,32]
    const float* __restrict__ conv_bias, // [32]
    float* __restrict__ g,               // [N,32] (ws)
    float* __restrict__ out,             // [N,32] (ws)
    int nNodes)
{
    __shared__ float xs[16 * 65];            // x tile, padded row stride 65
    __shared__ float w0[IN_FEAT * H_DIM];    // lin0_w staged
    __shared__ float hs[16][H_DIM + 1];      // h tile, padded

    const int lane  = threadIdx.x;           // 0..31 (one wave)
    const int m0    = blockIdx.x * 16;
    const int mrow  = lane & 15;
    const int khalf = (lane >> 4) << 1;      // 0 or 2
    const int rbase = (lane >> 4) * 8;

    // Stage x tile (coalesced, rows beyond N zero-filled)
    for (int i = lane; i < 16 * IN_FEAT; i += 32) {
        int row = i >> 6, col = i & 63;
        xs[row * 65 + col] =
            (m0 + row < nNodes) ? x[(size_t)(m0 + row) * IN_FEAT + col] : 0.0f;
    }
    // Stage lin0_w (coalesced)
    for (int i = lane; i < IN_FEAT * H_DIM; i += 32) w0[i] = lin0_w[i];
    __syncthreads();

    // ---- h = x_tile @ lin0_w : two 16x16 output tiles, K = 64 -------------
    v8f acc0 = {}, acc1 = {};
    #pragma unroll
    for (int k0 = 0; k0 < IN_FEAT; k0 += 4) {
        v2f a;
        a.x = xs[mrow * 65 + k0 + khalf];
        a.y = xs[mrow * 65 + k0 + khalf + 1];
        v2f b0 = { w0[(k0 + khalf) * H_DIM + mrow],
                   w0[(k0 + khalf + 1) * H_DIM + mrow] };
        v2f b1 = { w0[(k0 + khalf) * H_DIM + 16 + mrow],
                   w0[(k0 + khalf + 1) * H_DIM + 16 + mrow] };
        acc0 = __builtin_amdgcn_wmma_f32_16x16x4_f32(false, a, false, b0,
                                                     (short)0, acc0, false, false);
        acc1 = __builtin_amdgcn_wmma_f32_16x16x4_f32(false, a, false, b1,
                                                     (short)0, acc1, false, false);
    }

    const int nc0 = mrow, nc1 = 16 + mrow;
    const float lb0 = lin0_b[nc0], lb1 = lin0_b[nc1];
    #pragma unroll
    for (int v = 0; v < 8; ++v) {
        hs[rbase + v][nc0] = fmaxf(acc0[v] + lb0, 0.0f);
        hs[rbase + v][nc1] = fmaxf(acc1[v] + lb1, 0.0f);
    }
    __syncthreads();

    // ---- g = h @ W, r = h @ conv_root : K = 32 ----------------------------
    v8f ga0 = {}, ga1 = {}, ra0 = {}, ra1 = {};
    #pragma unroll
    for (int k0 = 0; k0 < H_DIM; k0 += 4) {
        v2f a;
        a.x = hs[mrow][k0 + khalf];
        a.y = hs[mrow][k0 + khalf + 1];
        v2f bw0 = { W[(k0 + khalf) * H_DIM + nc0],
                    W[(k0 + khalf + 1) * H_DIM + nc0] };
        v2f bw1 = { W[(k0 + khalf) * H_DIM + nc1],
                    W[(k0 + khalf + 1) * H_DIM + nc1] };
        v2f br0 = { conv_root[(k0 + khalf) * H_DIM + nc0],
                    conv_root[(k0 + khalf + 1) * H_DIM + nc0] };
        v2f br1 = { conv_root[(k0 + khalf) * H_DIM + nc1],
                    conv_root[(k0 + khalf + 1) * H_DIM + nc1] };
        ga0 = __builtin_amdgcn_wmma_f32_16x16x4_f32(false, a, false, bw0,
                                                    (short)0, ga0, false, false);
        ga1 = __builtin_amdgcn_wmma_f32_16x16x4_f32(false, a, false, bw1,
                                                    (short)0, ga1, false, false);
        ra0 = __builtin_amdgcn_wmma_f32_16x16x4_f32(false, a, false, br0,
                                                    (short)0, ra0, false, false);
        ra1 = __builtin_amdgcn_wmma_f32_16x16x4_f32(false, a, false, br1,
                                                    (short)0, ra1, false, false);
    }

    const float cb0 = conv_bias[nc0], cb1 = conv_bias[nc1];
    #pragma unroll
    for (int v = 0; v < 8; ++v) {
        int row = m0 + rbase + v;
        if (row < nNodes) {
            g[row * H_DIM + nc0]   = ga0[v];
            g[row * H_DIM + nc1]   = ga1[v];
            out[row * H_DIM + nc0] = ra0[v] + cb0;
            out[row * H_DIM + nc1] = ra1[v] + cb1;
        }
    }
}

// ---------------------------------------------------------------------------
// Kernel 3: scatter-add. One wave per edge, lane = channel.
//   out[dst, c] += g[src, c]   via global_atomic_add_f32 (no return).
// ---------------------------------------------------------------------------
__global__ __launch_bounds__(256) void scatter_kernel(
    const long long* __restrict__ edge_index, // [2, E] int64
    const float* __restrict__ g,
    float* __restrict__ out,
    int nEdges)
{
    int t = blockIdx.x * 256 + threadIdx.x;
    int e = t >> 5;
    int c = t & 31;
    if (e >= nEdges) return;
    int src = (int)edge_index[e];
    int dst = (int)edge_index[(size_t)nEdges + e];
    atomicAdd(&out[dst * H_DIM + c], g[src * H_DIM + c]);
}

// ---------------------------------------------------------------------------
// Kernel 4: per-edge scoring. One thread per edge.
//   emb = out[src] * out[dst];  p = relu(emb @ lin1_w + lin1_b);
//   score = p @ lin2_w + lin2_b
// out[] (6.4 MB) is L2-resident -> gathers served by the 192 MB L2.
// ---------------------------------------------------------------------------
__global__ __launch_bounds__(256) void edge_score_kernel(
    const long long* __restrict__ edge_index,
    const float* __restrict__ outn,
    const float* __restrict__ lin1_w,  // [32,8]
    const float* __restrict__ lin1_b,  // [8]
    const float* __restrict__ lin2_w,  // [8,1]
    const float* __restrict__ lin2_b,  // [1]
    float* __restrict__ score,
    int nEdges)
{
    __shared__ float l1w[H_DIM * 8];
    __shared__ float l1b[8];
    __shared__ float l2w[8];
    __shared__ float l2b0;
    int tid = threadIdx.x;
    l1w[tid] = lin1_w[tid];               // blockDim == 256 == 32*8
    if (tid < 8) { l1b[tid] = lin1_b[tid]; l2w[tid] = lin2_w[tid]; }
    if (tid == 0) l2b0 = lin2_b[0];
    __syncthreads();

    int e = blockIdx.x * 256 + tid;
    if (e >= nEdges) return;
    int src = (int)edge_index[e];
    int dst = (int)edge_index[(size_t)nEdges + e];

    const float4* ps = (const float4*)(outn + (size_t)src * H_DIM);
    const float4* pd = (const float4*)(outn + (size_t)dst * H_DIM);
    float emb[H_DIM];
    #pragma unroll
    for (int i = 0; i < 8; ++i) {
        float4 a = ps[i], b = pd[i];
        emb[i * 4 + 0] = a.x * b.x;
        emb[i * 4 + 1] = a.y * b.y;
        emb[i * 4 + 2] = a.z * b.z;
        emb[i * 4 + 3] = a.w * b.w;
    }
    float s = l2b0;
    #pragma unroll
    for (int j = 0; j < 8; ++j) {
        float p = l1b[j];
        #pragma unroll
        for (int c = 0; c < H_DIM; ++c)
            p = fmaf(emb[c], l1w[c * 8 + j], p);
        s = fmaf(fmaxf(p, 0.0f), l2w[j], s);
    }
    score[e] = s;
}

// ---------------------------------------------------------------------------
// Launcher. Workspace layout (floats): W[1024] | g[N*32] | out[N*32].
// ---------------------------------------------------------------------------
extern "C" void kernel_launch(void* const* d_in, const int* in_sizes, int n_in,
                              void* d_out, int out_size, void* d_ws, size_t ws_size,
                              hipStream_t stream) {
    const float*      x          = (const float*)d_in[0];
    const long long*  edge_index = (const long long*)d_in[1];   // int64 per reference
    const float*      lin0_w     = (const float*)d_in[2];
    const float*      lin0_b     = (const float*)d_in[3];
    const float*      nn_w1      = (const float*)d_in[4];
    const float*      nn_b1      = (const float*)d_in[5];
    const float*      nn_w2      = (const float*)d_in[6];
    const float*      nn_b2      = (const float*)d_in[7];
    const float*      conv_root  = (const float*)d_in[8];
    const float*      conv_bias  = (const float*)d_in[9];
    const float*      lin1_w     = (const float*)d_in[10];
    const float*      lin1_b     = (const float*)d_in[11];
    const float*      lin2_w     = (const float*)d_in[12];
    const float*      lin2_b     = (const float*)d_in[13];

    const int nNodes = in_sizes[0] / IN_FEAT;
    const int nEdges = in_sizes[1] / 2;

    float* Wg   = (float*)d_ws;
    float* g    = Wg + H_DIM * H_DIM;
    float* outn = g + (size_t)nNodes * H_DIM;

    wedge_kernel<<<1, 256, 0, stream>>>(nn_w1, nn_b1, nn_w2, nn_b2, Wg);

    const int nTiles = (nNodes + 15) / 16;
    node_kernel<<<nTiles, 32, 0, stream>>>(x, lin0_w, lin0_b, Wg, conv_root,
                                           conv_bias, g, outn, nNodes);

    const long long scatterThreads = (long long)nEdges * H_DIM;
    scatter_kernel<<<(int)((scatterThreads + 255) / 256), 256, 0, stream>>>(
        edge_index, g, outn, nEdges);

    edge_score_kernel<<<(nEdges + 255) / 256, 256, 0, stream>>>(
        edge_index, outn, lin1_w, lin1_b, lin2_w, lin2_b, (float*)d_out, nEdges);
}